// SimpleGCNEncoder_39178691674344
// MI455X (gfx1250) — compile-verified
//
#include <hip/hip_runtime.h>
#include <hip/hip_bf16.h>

// ---------------------------------------------------------------------------
// SimpleGCNEncoder for MI455X (gfx1250, wave32).
// Dense GEMMs use V_WMMA_F32_16X16X4_F32 (full fp32 precision; workload is
// L2-/atomic-bound so low-precision WMMA buys nothing). Edge aggregation uses
// non-returning global f32 atomics (RMW in L2; whole working set < 192MB L2).
// ---------------------------------------------------------------------------

typedef __attribute__((ext_vector_type(2))) float v2f;
typedef __attribute__((ext_vector_type(8))) float v8f;

#define D_DIM   64
#define HID_DIM 128

__device__ __forceinline__ v8f wmma_f32_k4(v2f a, v2f b, v8f c) {
  // 8 args: (neg_a, A, neg_b, B, c_mod, C, reuse_a, reuse_b)
  return __builtin_amdgcn_wmma_f32_16x16x4_f32(
      false, a, false, b, (short)0, c, false, false);
}

// ---------------------------------------------------------------------------
// small elementwise helpers
// ---------------------------------------------------------------------------
__global__ void k_fill(float* __restrict__ p, float v, int n) {
  int i = blockIdx.x * blockDim.x + threadIdx.x;
  if (i < n) p[i] = v;
}

__global__ void k_edge_deg(const int* __restrict__ ei, float* __restrict__ deg, int E) {
  int e = blockIdx.x * blockDim.x + threadIdx.x;
  if (e >= E) return;
  int dst = ei[E + e];
  atomicAdd(&deg[dst], 1.0f);
}

__global__ void k_rsqrt(const float* __restrict__ deg, float* __restrict__ dinv, int n) {
  int i = blockIdx.x * blockDim.x + threadIdx.x;
  if (i < n) dinv[i] = rsqrtf(deg[i]);
}

// agg[i][f] = hw[i][f] * dinv[i]^2   (self-loop term, also zero-initializes agg)
__global__ void k_selfinit(const float* __restrict__ hw, const float* __restrict__ dinv,
                           float* __restrict__ agg, int N) {
  int i = blockIdx.x * blockDim.x + threadIdx.x;
  int total = N * D_DIM;
  if (i >= total) return;
  int node = i >> 6;
  float di = dinv[node];
  agg[i] = hw[i] * (di * di);
}

// agg[dst] += hw[src] * dinv[src]*dinv[dst]; 16 threads per edge, float4 gather
__global__ void k_edge_scatter(const int* __restrict__ ei, const float* __restrict__ hw,
                               const float* __restrict__ dinv, float* __restrict__ agg, int E) {
  int tid = blockIdx.x * blockDim.x + threadIdx.x;
  if (tid >= E * 16) return;
  int e  = tid >> 4;
  int fq = (tid & 15) * 4;
  int src = ei[e];
  int dst = ei[E + e];
  float coef = dinv[src] * dinv[dst];
  const float4* row = reinterpret_cast<const float4*>(hw + (size_t)src * D_DIM);
  float4 v = row[tid & 15];
  float* outp = agg + (size_t)dst * D_DIM + fq;
  atomicAdd(outp + 0, v.x * coef);
  atomicAdd(outp + 1, v.y * coef);
  atomicAdd(outp + 2, v.z * coef);
  atomicAdd(outp + 3, v.w * coef);
}

// h[i][f] = relu(agg[i][f] + b[f])   (in place)
__global__ void k_bias_relu(float* __restrict__ h, const float* __restrict__ bias, int N) {
  int i = blockIdx.x * blockDim.x + threadIdx.x;
  int total = N * D_DIM;
  if (i >= total) return;
  float v = h[i] + bias[i & (D_DIM - 1)];
  h[i] = v > 0.0f ? v : 0.0f;
}

__global__ void k_count(const int* __restrict__ batch, float* __restrict__ cnt, int N) {
  int i = blockIdx.x * blockDim.x + threadIdx.x;
  if (i < N) atomicAdd(&cnt[batch[i]], 1.0f);
}

__global__ void k_mean(const float* __restrict__ pool, const float* __restrict__ cnt,
                       float* __restrict__ out, int G) {
  int i = blockIdx.x * blockDim.x + threadIdx.x;
  int total = G * HID_DIM;
  if (i >= total) return;
  int g = i >> 7;  // /128
  float c = cnt[g];
  out[i] = pool[i] / (c > 1.0f ? c : 1.0f);
}

// ---------------------------------------------------------------------------
// WMMA GEMM kernels. One wave32 owns a 16-node tile (EXEC all ones for WMMA).
// A-frag layout (ISA 7.12.2, 32-bit A 16x4): lane<16: M=lane, K={k,k+1};
//   lane>=16: M=lane-16, K={k+2,k+3}.  C/D: VGPR r -> M=r (lanes 0-15),
//   M=r+8 (lanes 16-31), N=lane&15.
// ---------------------------------------------------------------------------

// h0 = concat(emb[code], depth[dep]) @ projW(32x64) + projb
__global__ void k_embed_proj(const int* __restrict__ xn, const float* __restrict__ emb,
                             const float* __restrict__ dep, const float* __restrict__ W,
                             const float* __restrict__ bias, float* __restrict__ out, int N) {
  int lane = threadIdx.x & 31;
  int wave = threadIdx.x >> 5;
  int tile = blockIdx.x * (blockDim.x >> 5) + wave;
  int base = tile * 16;
  if (base >= N) return;
  int half = lane >> 4;
  int m    = lane & 15;
  int node = base + m;
  int code = xn[2 * node];
  int dpt  = xn[2 * node + 1];
  dpt = dpt < 0 ? 0 : (dpt > 19 ? 19 : dpt);
  const float* rowE = emb + code * 16;
  const float* rowD = dep + dpt * 16;
  for (int nt = 0; nt < 4; ++nt) {
    int n0 = nt * 16;
    v8f c = {0.f, 0.f, 0.f, 0.f, 0.f, 0.f, 0.f, 0.f};
    for (int k = 0; k < 32; k += 4) {
      int ka = k + 2 * half;          // even, never straddles the 16 boundary
      v2f a, b;
      a.x = (ka < 16) ? rowE[ka]     : rowD[ka - 16];
      a.y = (ka + 1 < 16) ? rowE[ka + 1] : rowD[ka - 15];
      b.x = W[ka * D_DIM + n0 + m];
      b.y = W[(ka + 1) * D_DIM + n0 + m];
      c = wmma_f32_k4(a, b, c);
    }
    float bv = bias[n0 + m];
    for (int r = 0; r < 8; ++r) {
      int row = base + r + 8 * half;
      out[(size_t)row * D_DIM + n0 + m] = c[r] + bv;
    }
  }
}

// hw = h(Nx64) @ W(64x64)  (no bias)
__global__ void k_gemm64(const float* __restrict__ h, const float* __restrict__ W,
                         float* __restrict__ out, int N) {
  int lane = threadIdx.x & 31;
  int wave = threadIdx.x >> 5;
  int tile = blockIdx.x * (blockDim.x >> 5) + wave;
  int base = tile * 16;
  if (base >= N) return;
  int half = lane >> 4;
  int m    = lane & 15;
  const float* A = h + (size_t)base * D_DIM;
  for (int nt = 0; nt < 4; ++nt) {
    int n0 = nt * 16;
    v8f c = {0.f, 0.f, 0.f, 0.f, 0.f, 0.f, 0.f, 0.f};
    for (int k = 0; k < D_DIM; k += 4) {
      int ka = k + 2 * half;
      v2f a, b;
      a.x = A[(size_t)m * D_DIM + ka];
      a.y = A[(size_t)m * D_DIM + ka + 1];
      b.x = W[ka * D_DIM + n0 + m];
      b.y = W[(ka + 1) * D_DIM + n0 + m];
      c = wmma_f32_k4(a, b, c);
    }
    for (int r = 0; r < 8; ++r) {
      int row = base + r + 8 * half;
      out[(size_t)row * D_DIM + n0 + m] = c[r];
    }
  }
}

// hfin = h(Nx64) @ finW(64x128) + finb, atomically pooled into pool[batch]
__global__ void k_final_pool(const float* __restrict__ h, const float* __restrict__ W,
                             const float* __restrict__ bias, const int* __restrict__ batch,
                             float* __restrict__ pool, int N) {
  int lane = threadIdx.x & 31;
  int wave = threadIdx.x >> 5;
  int tile = blockIdx.x * (blockDim.x >> 5) + wave;
  int base = tile * 16;
  if (base >= N) return;
  int half = lane >> 4;
  int m    = lane & 15;
  const float* A = h + (size_t)base * D_DIM;
  for (int nt = 0; nt < 8; ++nt) {
    int n0 = nt * 16;
    v8f c = {0.f, 0.f, 0.f, 0.f, 0.f, 0.f, 0.f, 0.f};
    for (int k = 0; k < D_DIM; k += 4) {
      int ka = k + 2 * half;
      v2f a, b;
      a.x = A[(size_t)m * D_DIM + ka];
      a.y = A[(size_t)m * D_DIM + ka + 1];
      b.x = W[ka * HID_DIM + n0 + m];
      b.y = W[(ka + 1) * HID_DIM + n0 + m];
      c = wmma_f32_k4(a, b, c);
    }
    float bv = bias[n0 + m];
    for (int r = 0; r < 8; ++r) {
      int row = base + r + 8 * half;
      int g = batch[row];
      atomicAdd(&pool[(size_t)g * HID_DIM + n0 + m], c[r] + bv);
    }
  }
}

// ---------------------------------------------------------------------------
static inline int cdiv(int a, int b) { return (a + b - 1) / b; }

extern "C" void kernel_launch(void* const* d_in, const int* in_sizes, int n_in,
                              void* d_out, int out_size, void* d_ws, size_t ws_size,
                              hipStream_t stream) {
  const int*   x_nodes = (const int*)d_in[0];
  const int*   edges   = (const int*)d_in[1];
  const int*   batch   = (const int*)d_in[2];
  const float* emb     = (const float*)d_in[3];
  const float* dep     = (const float*)d_in[4];
  const float* projW   = (const float*)d_in[5];
  const float* projB   = (const float*)d_in[6];
  const float* W1      = (const float*)d_in[7];
  const float* b1      = (const float*)d_in[8];
  const float* W2      = (const float*)d_in[9];
  const float* b2      = (const float*)d_in[10];
  const float* finW    = (const float*)d_in[11];
  const float* finB    = (const float*)d_in[12];
  float*       out     = (float*)d_out;

  const int N = in_sizes[0] / 2;
  const int E = in_sizes[1] / 2;
  const int G = out_size / HID_DIM;

  // workspace layout (floats)
  float* ws   = (float*)d_ws;
  float* bufA = ws;                              // N x 64 : h
  float* bufB = bufA + (size_t)N * D_DIM;        // N x 64 : hw
  float* deg  = bufB + (size_t)N * D_DIM;        // N
  float* dinv = deg + N;                         // N
  float* pool = dinv + N;                        // G x 128
  float* cnt  = pool + (size_t)G * HID_DIM;      // G

  const int B  = 256;                            // 8 wave32 per block
  const int gTiles = cdiv(cdiv(N, 16), 8);       // one 16-node tile per wave

  // degree & dinv
  k_fill<<<cdiv(N, B), B, 0, stream>>>(deg, 1.0f, N);
  k_edge_deg<<<cdiv(E, B), B, 0, stream>>>(edges, deg, E);
  k_rsqrt<<<cdiv(N, B), B, 0, stream>>>(deg, dinv, N);

  // embedding + projection  -> bufA
  k_embed_proj<<<gTiles, B, 0, stream>>>(x_nodes, emb, dep, projW, projB, bufA, N);

  // GCN layer 1: bufB = bufA@W1 ; bufA = relu(agg + self + b1)
  k_gemm64<<<gTiles, B, 0, stream>>>(bufA, W1, bufB, N);
  k_selfinit<<<cdiv(N * D_DIM, B), B, 0, stream>>>(bufB, dinv, bufA, N);
  k_edge_scatter<<<cdiv(E * 16, B), B, 0, stream>>>(edges, bufB, dinv, bufA, E);
  k_bias_relu<<<cdiv(N * D_DIM, B), B, 0, stream>>>(bufA, b1, N);

  // GCN layer 2
  k_gemm64<<<gTiles, B, 0, stream>>>(bufA, W2, bufB, N);
  k_selfinit<<<cdiv(N * D_DIM, B), B, 0, stream>>>(bufB, dinv, bufA, N);
  k_edge_scatter<<<cdiv(E * 16, B), B, 0, stream>>>(edges, bufB, dinv, bufA, E);
  k_bias_relu<<<cdiv(N * D_DIM, B), B, 0, stream>>>(bufA, b2, N);

  // final layer + mean pool
  k_fill<<<cdiv(G * HID_DIM, B), B, 0, stream>>>(pool, 0.0f, G * HID_DIM);
  k_fill<<<cdiv(G, B), B, 0, stream>>>(cnt, 0.0f, G);
  k_final_pool<<<gTiles, B, 0, stream>>>(bufA, finW, finB, batch, pool, N);
  k_count<<<cdiv(N, B), B, 0, stream>>>(batch, cnt, N);
  k_mean<<<cdiv(G * HID_DIM, B), B, 0, stream>>>(pool, cnt, out, G);
}